// DAGGenome_63737314673345
// MI455X (gfx1250) — compile-verified
//
#include <hip/hip_runtime.h>
#include <hip/hip_bf16.h>
#include <stdint.h>

// DAG reachability from node 0 (fixed-point BFS), N = 8192.
// Strategy: single WGP, everything in LDS.
//   - edges staged global->LDS with CDNA5 async DMA (global_load_async_to_lds_b128)
//   - reachability bitset (256 dwords) in LDS, set via ds_or_b32 atomics
//   - each thread owns 8 nodes, expands each node exactly once
//   - fixed point detected with a monotone LDS counter (race-free double barrier)

#define N_NODES 8192
#define NWORDS  (N_NODES / 32)   // 256 mask dwords
#define TPB     1024             // 32 wave32s on one WGP
#define NPT     (N_NODES / TPB)  // 8 nodes per thread

__global__ __launch_bounds__(TPB)
void dag_reach_kernel(const int* __restrict__ left,
                      const int* __restrict__ right,
                      float* __restrict__ out) {
    __shared__ int      sLeft[N_NODES];    // 32 KB
    __shared__ int      sRight[N_NODES];   // 32 KB
    __shared__ unsigned mask[NWORDS];      // 1 KB reachability bitset
    __shared__ unsigned expCount;          // monotone #expanded nodes

    const int tid = threadIdx.x;

    // ---- Stage edge arrays into LDS via async global->LDS (ASYNCcnt path).
    // 8192 ints = 2048 x 16B chunks per array; 1024 threads -> 2 chunks each.
#pragma unroll
    for (int j = 0; j < 2; ++j) {
        const int      chunk = tid + j * TPB;               // 16-byte chunk index
        const unsigned goff  = (unsigned)chunk * 16u;       // byte offset in global
        const unsigned loffL = (unsigned)(uintptr_t)(&sLeft[chunk * 4]);
        const unsigned loffR = (unsigned)(uintptr_t)(&sRight[chunk * 4]);
        asm volatile("global_load_async_to_lds_b128 %0, %1, %2"
                     :: "v"(loffL), "v"(goff), "s"(left) : "memory");
        asm volatile("global_load_async_to_lds_b128 %0, %1, %2"
                     :: "v"(loffR), "v"(goff), "s"(right) : "memory");
    }

    // Init bitset while the DMA is in flight.
    for (int i = tid; i < NWORDS; i += TPB) mask[i] = 0u;
    if (tid == 0) { expCount = 0u; mask[0] = 1u; }  // node 0 reachable

    asm volatile("s_wait_asynccnt 0" ::: "memory");  // drain this wave's async loads
    __syncthreads();                                  // edges + bitset visible to all

    // ---- Fixed-point BFS. Thread t owns nodes {t + j*1024 : j=0..7}.
    unsigned expanded = 0u;   // bit j: owned node j already expanded
    unsigned prev     = 0u;

    for (int iter = 0; iter <= N_NODES; ++iter) {
        unsigned newly = 0u;
        if (expanded != ((1u << NPT) - 1u)) {
#pragma unroll
            for (int j = 0; j < NPT; ++j) {
                const unsigned bit = 1u << j;
                if (expanded & bit) continue;
                const int node = tid + j * TPB;
                if (mask[node >> 5] & (1u << (node & 31))) {
                    const int l = sLeft[node];
                    const int r = sRight[node];
                    if (l >= 0) atomicOr(&mask[l >> 5], 1u << (l & 31));
                    if (r >= 0) atomicOr(&mask[r >> 5], 1u << (r & 31));
                    expanded |= bit;
                    ++newly;
                }
            }
        }
        if (newly) atomicAdd(&expCount, newly);

        __syncthreads();                 // B1: all expansions & count adds visible
        const unsigned cur = expCount;   // no writers between B1 and B2 -> uniform
        __syncthreads();                 // B2: all reads done before next iter's adds

        if (cur == prev) break;          // uniform: no new node expanded -> fixpoint
        prev = cur;
    }

    // ---- Emit mask as float 0/1 (reference output is bool -> float per harness).
#pragma unroll
    for (int j = 0; j < NPT; ++j) {
        const int node = tid + j * TPB;
        out[node] = ((mask[node >> 5] >> (node & 31)) & 1u) ? 1.0f : 0.0f;
    }
}

extern "C" void kernel_launch(void* const* d_in, const int* in_sizes, int n_in,
                              void* d_out, int out_size, void* d_ws, size_t ws_size,
                              hipStream_t stream) {
    // setup_inputs order: thresholds (unused by reference), left, right
    const int* left  = (const int*)d_in[1];
    const int* right = (const int*)d_in[2];
    float*     out   = (float*)d_out;

    (void)in_sizes; (void)n_in; (void)out_size; (void)d_ws; (void)ws_size;

    dag_reach_kernel<<<1, TPB, 0, stream>>>(left, right, out);
}